// PFNet_11304353923242
// MI455X (gfx1250) — compile-verified
//
#include <hip/hip_runtime.h>
#include <hip/hip_bf16.h>
#include <math.h>

typedef __attribute__((ext_vector_type(2))) float v2f;
typedef __attribute__((ext_vector_type(8))) float v8f;

#define NEG_ATT 0.2f
#define NEG_ACT 0.01f
#define SCAN_T 1024

__device__ __forceinline__ float leakyf(float x, float s) { return x >= 0.f ? x : x * s; }

__device__ __forceinline__ void atomicMaxF(float* addr, float v) {
    if (v >= 0.f) atomicMax((int*)addr, __float_as_int(v));
    else          atomicMin((unsigned int*)addr, __float_as_uint(v));
}

// ---------------------------------------------------------------------------
// fp32 WMMA GEMM:  C[M,Ncol] = leaky(A[M,K] @ B[K,Ncol] + bias, slope)
//
// Launcher guarantees: M % 16 == 0, Ncol % 16 == 0, K % 4 == 0 (template).
// One wave per 16-col panel; each wave computes MT consecutive 16x16 tiles
// down M, reusing the register-resident B panel. No per-lane guards ->
// EXEC all-1s through every WMMA. All index math is 32-bit (extents < 2^31).
//
// VGPR layouts per CDNA5 ISA (16x16x4 f32):
//   A: lane(lo=lane&15,hi=lane>>4) holds A[lo][k0+2hi], A[lo][k0+2hi+1]
//   B: lane holds B[k0+2hi][lo], B[k0+2hi+1][lo]
//   D: vgpr r holds D[r + 8*hi][lo]
// ---------------------------------------------------------------------------
template <int K, int MT>
__global__ void wmma_gemm_f32(const float* __restrict__ A, const float* __restrict__ B,
                              const float* __restrict__ bias, float* __restrict__ C,
                              int M, int ldb, float slope) {
    const int lane = threadIdx.x;   // 0..31
    const int hi   = lane >> 4;
    const int lo   = lane & 15;
    const int colBase = blockIdx.y * 16;
    const int col  = colBase + lo;

    // Preload this wave's B panel (K x 16) into registers, WMMA-layout.
    v2f breg[K / 4];
#pragma unroll
    for (int kk = 0; kk < K / 4; ++kk) {
        const int ka = 4 * kk + 2 * hi;
        breg[kk].x = B[ka * ldb + col];
        breg[kk].y = B[(ka + 1) * ldb + col];
    }
    const float bv = bias[col];

#pragma unroll
    for (int t = 0; t < MT; ++t) {
        const int row0 = (blockIdx.x * MT + t) * 16;   // uniform (scalar) value
        if (row0 >= M) break;                          // uniform branch, EXEC untouched

        const float* arow = A + (row0 + lo) * K + 2 * hi;
        v8f acc = {0.f, 0.f, 0.f, 0.f, 0.f, 0.f, 0.f, 0.f};
#pragma unroll
        for (int kk = 0; kk < K / 4; ++kk) {
            v2f a = *(const v2f*)(arow + 4 * kk);      // contiguous pair -> b64 load
            acc = __builtin_amdgcn_wmma_f32_16x16x4_f32(false, a, false, breg[kk],
                                                        (short)0, acc, false, false);
        }
        const int cbase = row0 * ldb + colBase + lo;
#pragma unroll
        for (int r = 0; r < 8; ++r) {
            float v = acc[r] + bv;
            C[cbase + (r + 8 * hi) * ldb] = leakyf(v, slope);
        }
    }
}

// ---------------------------------------------------------------------------
// Small helper kernels
// ---------------------------------------------------------------------------
__global__ void fill_k(float* __restrict__ p, float v, int n) {
    int i = blockIdx.x * blockDim.x + threadIdx.x;
    if (i < n) p[i] = v;
}

__global__ void build_xpad_k(const float* __restrict__ x, float* __restrict__ xp, int N) {
    int i = blockIdx.x * blockDim.x + threadIdx.x;
    if (i >= N) return;
    xp[i * 4 + 0] = x[i * 3 + 0];
    xp[i * 4 + 1] = x[i * 3 + 1];
    xp[i * 4 + 2] = x[i * 3 + 2];
    xp[i * 4 + 3] = 0.f;
}

// zero-pad both dims: [Kin,Cin] -> [Kpad,Cpad]
__global__ void build_wpad_k(const float* __restrict__ W, float* __restrict__ Wp,
                             int Kin, int Cin, int Kpad, int Cpad) {
    int i = blockIdx.x * blockDim.x + threadIdx.x;
    if (i >= Kpad * Cpad) return;
    int r = i / Cpad, c = i % Cpad;
    Wp[i] = (r < Kin && c < Cin) ? W[r * Cin + c] : 0.f;
}

__global__ void build_bpad_k(const float* __restrict__ b, float* __restrict__ bp,
                             int nIn, int nPad) {
    int i = blockIdx.x * blockDim.x + threadIdx.x;
    if (i < nPad) bp[i] = (i < nIn) ? b[i] : 0.f;
}

// hcat[N,36] = [x(3) | x3(32) | 0]
__global__ void build_hcat_k(const float* __restrict__ x, const float* __restrict__ x3,
                             float* __restrict__ hc, int N) {
    int i = blockIdx.x * blockDim.x + threadIdx.x;
    if (i >= N) return;
    float* o = hc + i * 36;
    o[0] = x[i * 3 + 0];
    o[1] = x[i * 3 + 1];
    o[2] = x[i * 3 + 2];
    const float* a = x3 + i * 32;
#pragma unroll
    for (int j = 0; j < 32; ++j) o[3 + j] = a[j];
    o[35] = 0.f;
}

// a_s[n,h] = sum_f h[n,h,f]*att_src[h,f]   (and same for a_d)
__global__ void attn_coef_k(const float* __restrict__ h, const float* __restrict__ atts,
                            const float* __restrict__ attd, float* __restrict__ as_,
                            float* __restrict__ ad_, int N) {
    int idx = blockIdx.x * blockDim.x + threadIdx.x;
    if (idx >= N * 4) return;
    int n = idx >> 2, hd = idx & 3;
    const float* hr = h + n * 128 + hd * 32;
    const float* s  = atts + hd * 32;
    const float* d  = attd + hd * 32;
    float vs = 0.f, vd = 0.f;
#pragma unroll
    for (int f = 0; f < 32; ++f) { float t = hr[f]; vs += t * s[f]; vd += t * d[f]; }
    as_[idx] = vs;
    ad_[idx] = vd;
}

// ---------------------------------------------------------------------------
// Edge kernels (edge list = [src(E0) | dst(E0)] + implicit self loops)
// ---------------------------------------------------------------------------
__global__ void edge_max_k(const int* __restrict__ eidx, int E0, int N,
                           const float* __restrict__ as_, const float* __restrict__ ad_,
                           float* __restrict__ mmax) {
    int e = blockIdx.x * blockDim.x + threadIdx.x;
    if (e >= E0 + N) return;
    int s = (e < E0) ? eidx[e]      : (e - E0);
    int d = (e < E0) ? eidx[E0 + e] : (e - E0);
#pragma unroll
    for (int hd = 0; hd < 4; ++hd) {
        float ev = leakyf(as_[s * 4 + hd] + ad_[d * 4 + hd], NEG_ATT);
        atomicMaxF(&mmax[d * 4 + hd], ev);
    }
}

__global__ void edge_expsum_k(const int* __restrict__ eidx, int E0, int N,
                              const float* __restrict__ as_, const float* __restrict__ ad_,
                              const float* __restrict__ mmax, float* __restrict__ ex,
                              float* __restrict__ denom) {
    int e = blockIdx.x * blockDim.x + threadIdx.x;
    if (e >= E0 + N) return;
    int s = (e < E0) ? eidx[e]      : (e - E0);
    int d = (e < E0) ? eidx[E0 + e] : (e - E0);
#pragma unroll
    for (int hd = 0; hd < 4; ++hd) {
        float ev = leakyf(as_[s * 4 + hd] + ad_[d * 4 + hd], NEG_ATT);
        float v  = expf(ev - mmax[d * 4 + hd]);
        ex[e * 4 + hd] = v;
        atomicAdd(&denom[d * 4 + hd], v);
    }
}

// accum[dst, h, f] += alpha * h[src, h, f]  -- one thread per (edge, f)
__global__ void edge_scatter_k(const int* __restrict__ eidx, int E0, int N,
                               const float* __restrict__ ex, const float* __restrict__ denom,
                               const float* __restrict__ h, float* __restrict__ accum) {
    long idx = (long)blockIdx.x * blockDim.x + threadIdx.x;
    long tot = (long)(E0 + N) * 32;
    if (idx >= tot) return;
    int e = (int)(idx >> 5);
    int f = (int)(idx & 31);
    int s = (e < E0) ? eidx[e]      : (e - E0);
    int d = (e < E0) ? eidx[E0 + e] : (e - E0);
#pragma unroll
    for (int hd = 0; hd < 4; ++hd) {
        float alpha = ex[e * 4 + hd] / (denom[d * 4 + hd] + 1e-16f);
        atomicAdd(&accum[d * 128 + hd * 32 + f], alpha * h[s * 128 + hd * 32 + f]);
    }
}

// xout[n,f] = leaky( mean_h accum[n,h,f] + bias[f], 0.01 )
__global__ void head_mean_k(const float* __restrict__ accum, const float* __restrict__ bias,
                            float* __restrict__ xout, int N) {
    int idx = blockIdx.x * blockDim.x + threadIdx.x;
    if (idx >= N * 32) return;
    int n = idx >> 5, f = idx & 31;
    const float* a = accum + n * 128;
    float v = 0.25f * (a[f] + a[32 + f] + a[64 + f] + a[96 + f]) + bias[f];
    xout[idx] = leakyf(v, NEG_ACT);
}

// ---------------------------------------------------------------------------
// Order-preserving compaction: exclusive scan of tower mask (x[:,0]==0)
// ---------------------------------------------------------------------------
__global__ void scan_local_k(const float* __restrict__ x, int N,
                             int* __restrict__ localRank, int* __restrict__ blockSums) {
    __shared__ int sh[SCAN_T];
    int t = threadIdx.x;
    int i = blockIdx.x * SCAN_T + t;
    int m = (i < N && x[i * 3] == 0.f) ? 1 : 0;
    sh[t] = m;
    __syncthreads();
    for (int off = 1; off < SCAN_T; off <<= 1) {
        int v = (t >= off) ? sh[t - off] : 0;
        __syncthreads();
        sh[t] += v;
        __syncthreads();
    }
    if (i < N) localRank[i] = sh[t] - m;   // exclusive rank in block
    if (t == SCAN_T - 1) blockSums[blockIdx.x] = sh[t];
}

__global__ void scan_blocks_k(int* __restrict__ blockSums, int nB, int* __restrict__ totalT) {
    if (blockIdx.x == 0 && threadIdx.x == 0) {
        int acc = 0;
        for (int b = 0; b < nB; ++b) { int v = blockSums[b]; blockSums[b] = acc; acc += v; }
        *totalT = acc;
    }
}

// out = [ct[:, :9] | ck[:, :9] | ct[:, 9:] | ck[:, 9:]] flattened; cands stride 16
__global__ void scatter_out_k(const float* __restrict__ cands, const float* __restrict__ x,
                              const int* __restrict__ localRank, const int* __restrict__ blockOff,
                              const int* __restrict__ totalT, float* __restrict__ out, int N) {
    int i = blockIdx.x * blockDim.x + threadIdx.x;
    if (i >= N) return;
    int T  = *totalT;
    int tr = blockOff[i / SCAN_T] + localRank[i];    // # towers before i
    bool tower = (x[i * 3] == 0.f);
    const float* row = cands + i * 16;
    if (tower) {
        float* o9 = out + tr * 9;
        for (int j = 0; j < 9; ++j) o9[j] = row[j];
        float* o3 = out + N * 9 + tr * 3;
        for (int j = 0; j < 3; ++j) o3[j] = row[9 + j];
    } else {
        int kr = i - tr;
        float* o9 = out + T * 9 + kr * 9;
        for (int j = 0; j < 9; ++j) o9[j] = row[j];
        float* o3 = out + N * 9 + T * 3 + kr * 3;
        for (int j = 0; j < 3; ++j) o3[j] = row[9 + j];
    }
}

// ---------------------------------------------------------------------------
// Host orchestration
// ---------------------------------------------------------------------------
extern "C" void kernel_launch(void* const* d_in, const int* in_sizes, int n_in,
                              void* d_out, int out_size, void* d_ws, size_t ws_size,
                              hipStream_t stream) {
    const float* x    = (const float*)d_in[0];
    const int*   eidx = (const int*)d_in[1];    // [2,E] row-major: [src | dst]
    const float* W1 = (const float*)d_in[3],  *as1 = (const float*)d_in[4];
    const float* ad1 = (const float*)d_in[5], *b1  = (const float*)d_in[6];
    const float* W2 = (const float*)d_in[7],  *as2 = (const float*)d_in[8];
    const float* ad2 = (const float*)d_in[9], *b2  = (const float*)d_in[10];
    const float* W3 = (const float*)d_in[11], *as3 = (const float*)d_in[12];
    const float* ad3 = (const float*)d_in[13],*b3  = (const float*)d_in[14];
    const float* mW1 = (const float*)d_in[15], *mb1 = (const float*)d_in[16];
    const float* mW2 = (const float*)d_in[17], *mb2 = (const float*)d_in[18];
    const float* mW3 = (const float*)d_in[19], *mb3 = (const float*)d_in[20];
    const float* mW4 = (const float*)d_in[21], *mb4 = (const float*)d_in[22];
    const float* mW5 = (const float*)d_in[23], *mb5 = (const float*)d_in[24];

    const int N    = in_sizes[0] / 3;           // multiple of 16 (100000)
    const int E0   = in_sizes[1] / 2;
    const int Etot = E0 + N;

    char* base = (char*)d_ws;
    size_t off = 0;
    auto alloc = [&](size_t nbytes) -> void* {
        void* p = base + off;
        off = (off + nbytes + 255) & ~(size_t)255;
        return p;
    };

    float* xpad  = (float*)alloc((size_t)N * 4    * sizeof(float));
    float* W1p   = (float*)alloc(4 * 128          * sizeof(float));
    float* h     = (float*)alloc((size_t)N * 128  * sizeof(float));
    float* as_   = (float*)alloc((size_t)N * 4    * sizeof(float));
    float* ad_   = (float*)alloc((size_t)N * 4    * sizeof(float));
    float* mmax  = (float*)alloc((size_t)N * 4    * sizeof(float));
    float* denom = (float*)alloc((size_t)N * 4    * sizeof(float));
    float* ex    = (float*)alloc((size_t)Etot * 4 * sizeof(float));
    float* accum = (float*)alloc((size_t)N * 128  * sizeof(float));
    float* xa    = (float*)alloc((size_t)N * 32   * sizeof(float));
    float* xb    = (float*)alloc((size_t)N * 32   * sizeof(float));
    float* cands = (float*)alloc((size_t)N * 16   * sizeof(float));
    float* Wp1   = (float*)alloc(36 * 32          * sizeof(float));
    float* W5p   = (float*)alloc(32 * 16          * sizeof(float));
    float* mb5p  = (float*)alloc(16               * sizeof(float));
    float* zbias = (float*)alloc(128              * sizeof(float));
    int*   localRank = (int*)alloc((size_t)N * sizeof(int));
    int*   blockOff  = (int*)alloc(1024 * sizeof(int));
    int*   totalT    = (int*)alloc(256);

    // aliases (regions dead at MLP time)
    float* hcat  = h;                         // N*36 <= N*128
    float* m1    = accum;                     // accum free after layer 3
    float* m2    = accum + (size_t)N * 32;
    float* m3    = accum + (size_t)N * 64;
    float* m4    = accum + (size_t)N * 96;

    const int TPB = 256;
    const int MT  = 4;
    const int gMT = (N / 16 + MT - 1) / MT;   // M-tile groups per GEMM

    fill_k<<<1, 128, 0, stream>>>(zbias, 0.f, 128);

    auto gat = [&](const float* Xin, bool firstLayer, const float* W, const float* atts,
                   const float* attd, const float* bias, float* Xout) {
        dim3 g1(gMT, 8);
        if (firstLayer)
            wmma_gemm_f32<4,  MT><<<g1, 32, 0, stream>>>(Xin, W, zbias, h, N, 128, 1.0f);
        else
            wmma_gemm_f32<32, MT><<<g1, 32, 0, stream>>>(Xin, W, zbias, h, N, 128, 1.0f);
        attn_coef_k<<<(N * 4 + TPB - 1) / TPB, TPB, 0, stream>>>(h, atts, attd, as_, ad_, N);
        fill_k<<<(N * 4 + TPB - 1) / TPB, TPB, 0, stream>>>(mmax, -INFINITY, N * 4);
        fill_k<<<(N * 4 + TPB - 1) / TPB, TPB, 0, stream>>>(denom, 0.f, N * 4);
        fill_k<<<(N * 128 + TPB - 1) / TPB, TPB, 0, stream>>>(accum, 0.f, N * 128);
        edge_max_k<<<(Etot + TPB - 1) / TPB, TPB, 0, stream>>>(eidx, E0, N, as_, ad_, mmax);
        edge_expsum_k<<<(Etot + TPB - 1) / TPB, TPB, 0, stream>>>(eidx, E0, N, as_, ad_,
                                                                  mmax, ex, denom);
        long tot = (long)Etot * 32;
        edge_scatter_k<<<(unsigned)((tot + TPB - 1) / TPB), TPB, 0, stream>>>(
            eidx, E0, N, ex, denom, h, accum);
        head_mean_k<<<(N * 32 + TPB - 1) / TPB, TPB, 0, stream>>>(accum, bias, Xout, N);
    };

    // --- GAT stack (input padded to K=4 so layer 1 also runs on WMMA) ---
    build_xpad_k<<<(N + TPB - 1) / TPB, TPB, 0, stream>>>(x, xpad, N);
    build_wpad_k<<<(4 * 128 + TPB - 1) / TPB, TPB, 0, stream>>>(W1, W1p, 3, 128, 4, 128);
    gat(xpad, true,  W1p, as1, ad1, b1, xa);  // x1 (uses zero-padded 4x128 weights)
    gat(xa,   false, W2,  as2, ad2, b2, xb);  // x2
    gat(xb,   false, W3,  as3, ad3, b3, xa);  // x3

    // --- MLP head (all WMMA, fused bias+leaky epilogue) ---
    build_hcat_k<<<(N + TPB - 1) / TPB, TPB, 0, stream>>>(x, xa, hcat, N);
    build_wpad_k<<<(36 * 32 + TPB - 1) / TPB, TPB, 0, stream>>>(mW1, Wp1, 35, 32, 36, 32);
    build_wpad_k<<<(32 * 16 + TPB - 1) / TPB, TPB, 0, stream>>>(mW5, W5p, 32, 12, 32, 16);
    build_bpad_k<<<1, 32, 0, stream>>>(mb5, mb5p, 12, 16);

    dim3 gm(gMT, 2), gc(gMT, 1);
    wmma_gemm_f32<36, MT><<<gm, 32, 0, stream>>>(hcat, Wp1, mb1,  m1,    N, 32, NEG_ACT);
    wmma_gemm_f32<32, MT><<<gm, 32, 0, stream>>>(m1,   mW2, mb2,  m2,    N, 32, NEG_ACT);
    wmma_gemm_f32<32, MT><<<gm, 32, 0, stream>>>(m2,   mW3, mb3,  m3,    N, 32, NEG_ACT);
    wmma_gemm_f32<32, MT><<<gm, 32, 0, stream>>>(m3,   mW4, mb4,  m4,    N, 32, NEG_ACT);
    wmma_gemm_f32<32, MT><<<gc, 32, 0, stream>>>(m4,   W5p, mb5p, cands, N, 16, 1.0f);

    // --- order-preserving tower/non-tower split ---
    int nB = (N + SCAN_T - 1) / SCAN_T;
    scan_local_k<<<nB, SCAN_T, 0, stream>>>(x, N, localRank, blockOff);
    scan_blocks_k<<<1, 32, 0, stream>>>(blockOff, nB, totalT);
    scatter_out_k<<<(N + TPB - 1) / TPB, TPB, 0, stream>>>(cands, x, localRank, blockOff,
                                                           totalT, (float*)d_out, N);
}